// GAT_Res_53008486367316
// MI455X (gfx1250) — compile-verified
//
#include <hip/hip_runtime.h>
#include <math.h>

// Model constants (from reference)
#define IN_DIM  256
#define HEADS   4
#define HID     128
#define DHID    512      // HEADS*HID
#define OUTD    128
#define NGRAPH  64
#define NLAYERS 4

// GEMM tiling
#define MBLK 128         // rows per block (8 waves x 16)
#define NBLK 64          // cols per block
#define KB   16          // k-chunk staged in LDS
#define ASTR (KB + 2)    // padded LDS row stride (dwords) for A  (TDM pad: 2 per 16)
#define BSTR (NBLK + 2)  // padded LDS row stride (dwords) for B  (TDM pad: 2 per 64)

typedef __attribute__((ext_vector_type(2))) float    v2f;
typedef __attribute__((ext_vector_type(8))) float    v8f;
typedef __attribute__((ext_vector_type(4))) unsigned v4u;
typedef __attribute__((ext_vector_type(8))) int      v8i;
typedef __attribute__((ext_vector_type(4))) int      v4i;

// ---------------------------------------------------------------------------
// TDM: 2D tile load global->LDS. D# per CDNA5 ISA ch.8 (groups 0/1; groups 2/3
// zero => 2D). pad_enable inserts LDS padding (bank-conflict avoidance).
//  pad_icode: pad_interval code (3 => every 16 dwords, 5 => every 64 dwords)
//  pad_acode: pad_amount  code (1 => 2 dwords)
// ---------------------------------------------------------------------------
__device__ __forceinline__ void tdm_load_2d(unsigned lds_off, const float* gptr,
                                            unsigned tile_x, unsigned tile_y,
                                            unsigned dim_x, unsigned dim_y,
                                            unsigned long long stride_y,
                                            unsigned pad_icode, unsigned pad_acode) {
  unsigned long long ga = (unsigned long long)gptr;
  v4u g0;
  g0[0] = 1u;                                            // count=1, user mode
  g0[1] = lds_off;                                       // lds_addr (bytes)
  g0[2] = (unsigned)(ga & 0xFFFFFFFFu);                  // global_addr[31:0]
  g0[3] = (unsigned)((ga >> 32) & 0x01FFFFFFu)           // global_addr[56:32]
        | (2u << 30);                                    // type=2 ("image")
  v8i g1;
  g1[0] = (int)((2u << 16)                               // data_size=4B
        | (1u << 20)                                     // pad_enable
        | (pad_icode << 22) | (pad_acode << 25));        // pad interval/amount
  g1[1] = (int)((dim_x & 0xFFFFu) << 16);                // tensor_dim0[15:0] (atomic addr=0)
  g1[2] = (int)(((dim_x >> 16) & 0xFFFFu) | ((dim_y & 0xFFFFu) << 16));
  g1[3] = (int)(((dim_y >> 16) & 0xFFFFu) | (tile_x << 16));   // tile_dim0
  g1[4] = (int)(tile_y & 0xFFFFu);                       // tile_dim1 (tile_dim2=0 => 2D)
  g1[5] = (int)(stride_y & 0xFFFFFFFFu);                 // tensor_dim0_stride[31:0]
  g1[6] = (int)((stride_y >> 32) & 0xFFFFu);             // stride[47:32] (dim1_stride=0)
  g1[7] = 0;
  v4i z4 = {};
#if __has_include(<hip/amd_detail/amd_gfx1250_TDM.h>)
  v8i z8 = {};
  __builtin_amdgcn_tensor_load_to_lds(g0, g1, z4, z4, z8, 0);  // clang-23 arity
#else
  __builtin_amdgcn_tensor_load_to_lds(g0, g1, z4, z4, 0);      // ROCm 7.2 arity
#endif
}

// ---------------------------------------------------------------------------
// Block-tiled GEMM: C[M x Nc] = A[M x K] @ B[K x Nc]
// 256 threads = 8 waves; wave w -> rows [w*16, w*16+16) of a 128x64 tile.
// A/B k-panels double-buffered in LDS via TDM; WMMA F32 16x16x4 from ds loads.
// TDM OOB reads (rows >= M in tail blocks) return zero -> safe accumulation.
// ---------------------------------------------------------------------------
__global__ void __launch_bounds__(256)
gemm_wmma_tdm(const float* __restrict__ A, const float* __restrict__ B,
              float* __restrict__ C, int M, int K, int Nc) {
  __shared__ float Alds[2][MBLK * ASTR];
  __shared__ float Blds[2][KB * BSTR];

  const int wid  = threadIdx.x >> 5;
  const int lane = threadIdx.x & 31;
  const int r    = lane & 15;
  const int half = lane >> 4;
  const int row0 = blockIdx.x * MBLK;
  const int col0 = blockIdx.y * NBLK;

  const unsigned aoff0 = (unsigned)(unsigned long long)(&Alds[0][0]);
  const unsigned aoff1 = (unsigned)(unsigned long long)(&Alds[1][0]);
  const unsigned boff0 = (unsigned)(unsigned long long)(&Blds[0][0]);
  const unsigned boff1 = (unsigned)(unsigned long long)(&Blds[1][0]);

  const unsigned rows_left = (unsigned)(M - row0);        // tail: OOB rows -> zeros
  const int nchunk = K / KB;

  if (wid == 0) {   // preload chunk 0 into buffer 0
    tdm_load_2d(aoff0, A + (size_t)row0 * K, KB, MBLK,
                (unsigned)K, rows_left, (unsigned long long)K, 3u, 1u);
    tdm_load_2d(boff0, B + (size_t)col0, NBLK, KB,
                (unsigned)(Nc - col0), (unsigned)K, (unsigned long long)Nc, 5u, 1u);
    __builtin_amdgcn_s_wait_tensorcnt(0);
  }
  __syncthreads();

  v8f acc0 = {}, acc1 = {}, acc2 = {}, acc3 = {};
  int cur = 0;
  for (int t = 0; t < nchunk; ++t) {
    const bool more = (t + 1 < nchunk);
    if (wid == 0 && more) {           // prefetch next chunk into other buffer
      const int k1 = (t + 1) * KB;
      tdm_load_2d(cur ? aoff0 : aoff1, A + (size_t)row0 * K + k1, KB, MBLK,
                  (unsigned)(K - k1), rows_left, (unsigned long long)K, 3u, 1u);
      tdm_load_2d(cur ? boff0 : boff1, B + (size_t)k1 * Nc + col0, NBLK, KB,
                  (unsigned)(Nc - col0), (unsigned)(K - k1), (unsigned long long)Nc, 5u, 1u);
    }

    const float* Ab = &Alds[cur][0];
    const float* Bb = &Blds[cur][0];
#pragma unroll
    for (int kk = 0; kk < KB; kk += 4) {
      const float* ar = Ab + (wid * 16 + r) * ASTR + kk + 2 * half;
      v2f a; a.x = ar[0]; a.y = ar[1];
      const float* b0p = Bb + (kk + 2 * half) * BSTR + r;
      const float* b1p = b0p + BSTR;
      v2f b0, b1, b2, b3;
      b0.x = b0p[0];  b0.y = b1p[0];
      b1.x = b0p[16]; b1.y = b1p[16];
      b2.x = b0p[32]; b2.y = b1p[32];
      b3.x = b0p[48]; b3.y = b1p[48];
      acc0 = __builtin_amdgcn_wmma_f32_16x16x4_f32(false, a, false, b0, (short)0, acc0, false, false);
      acc1 = __builtin_amdgcn_wmma_f32_16x16x4_f32(false, a, false, b1, (short)0, acc1, false, false);
      acc2 = __builtin_amdgcn_wmma_f32_16x16x4_f32(false, a, false, b2, (short)0, acc2, false, false);
      acc3 = __builtin_amdgcn_wmma_f32_16x16x4_f32(false, a, false, b3, (short)0, acc3, false, false);
    }

    if (wid == 0 && more) __builtin_amdgcn_s_wait_tensorcnt(0);
    __syncthreads();
    cur ^= 1;
  }

#pragma unroll
  for (int v = 0; v < 8; ++v) {
    int row = row0 + wid * 16 + v + 8 * half;
    if (row < M) {
      float* Cr = C + (size_t)row * Nc + col0 + r;
      Cr[0]  = acc0[v];
      Cr[16] = acc1[v];
      Cr[32] = acc2[v];
      Cr[48] = acc3[v];
    }
  }
}

// ---------------------------------------------------------------------------
// Small GEMM (wave-level, direct global loads) for the 64x128 output projection.
// ---------------------------------------------------------------------------
__global__ void gemm_wmma_f32(const float* __restrict__ A, const float* __restrict__ B,
                              const float* __restrict__ bias, float* __restrict__ C,
                              int M, int K, int Nc) {
  const int lane = threadIdx.x & 31;
  const int r    = lane & 15;
  const int half = lane >> 4;
  const int row0 = blockIdx.x * 16;
  const int col0 = blockIdx.y * 64;

  int arow = row0 + r; if (arow > M - 1) arow = M - 1;
  const float* Arow = A + (size_t)arow * K;

  v8f acc0 = {}, acc1 = {}, acc2 = {}, acc3 = {};
  for (int k0 = 0; k0 < K; k0 += 4) {
    v2f a;
    a.x = Arow[k0 + 2 * half];
    a.y = Arow[k0 + 2 * half + 1];
    const float* B0 = B + (size_t)(k0 + 2 * half) * Nc + col0 + r;
    const float* B1 = B0 + Nc;
    v2f b0, b1, b2, b3;
    b0.x = B0[0];  b0.y = B1[0];
    b1.x = B0[16]; b1.y = B1[16];
    b2.x = B0[32]; b2.y = B1[32];
    b3.x = B0[48]; b3.y = B1[48];
    acc0 = __builtin_amdgcn_wmma_f32_16x16x4_f32(false, a, false, b0, (short)0, acc0, false, false);
    acc1 = __builtin_amdgcn_wmma_f32_16x16x4_f32(false, a, false, b1, (short)0, acc1, false, false);
    acc2 = __builtin_amdgcn_wmma_f32_16x16x4_f32(false, a, false, b2, (short)0, acc2, false, false);
    acc3 = __builtin_amdgcn_wmma_f32_16x16x4_f32(false, a, false, b3, (short)0, acc3, false, false);
  }

  float bv0 = bias ? bias[col0 + r]      : 0.f;
  float bv1 = bias ? bias[col0 + 16 + r] : 0.f;
  float bv2 = bias ? bias[col0 + 32 + r] : 0.f;
  float bv3 = bias ? bias[col0 + 48 + r] : 0.f;

#pragma unroll
  for (int v = 0; v < 8; ++v) {
    int row = row0 + v + 8 * half;
    if (row < M) {
      float* Cr = C + (size_t)row * Nc + col0 + r;
      Cr[0]  = acc0[v] + bv0;
      Cr[16] = acc1[v] + bv1;
      Cr[32] = acc2[v] + bv2;
      Cr[48] = acc3[v] + bv3;
    }
  }
}

// ---------------------------------------------------------------------------
// Per-node attention scores: a_src[n,h] = <h[n,h,:], att_src[h,:]> (same for dst)
// ---------------------------------------------------------------------------
__global__ void attn_scores(const float* __restrict__ h, const float* __restrict__ att_s,
                            const float* __restrict__ att_d,
                            float* __restrict__ asr, float* __restrict__ adr) {
  const int n    = blockIdx.x;
  const int head = threadIdx.x >> 5;
  const int lane = threadIdx.x & 31;
  const float4 hv = *(const float4*)(h + (size_t)n * DHID + head * HID + lane * 4);
  const float4 s4 = *(const float4*)(att_s + head * HID + lane * 4);
  const float4 d4 = *(const float4*)(att_d + head * HID + lane * 4);
  float ss = hv.x * s4.x + hv.y * s4.y + hv.z * s4.z + hv.w * s4.w;
  float dd = hv.x * d4.x + hv.y * d4.y + hv.z * d4.z + hv.w * d4.w;
#pragma unroll
  for (int off = 16; off > 0; off >>= 1) {
    ss += __shfl_xor(ss, off, 32);
    dd += __shfl_xor(dd, off, 32);
  }
  if (lane == 0) {
    asr[n * HEADS + head] = ss;
    adr[n * HEADS + head] = dd;
  }
}

// Monotone float<->uint encoding so atomicMax(uint) implements float max.
__device__ __forceinline__ unsigned fenc(float f) {
  unsigned u = __float_as_uint(f);
  return (u & 0x80000000u) ? ~u : (u | 0x80000000u);
}
__device__ __forceinline__ float fdec(unsigned u) {
  unsigned b = (u & 0x80000000u) ? (u ^ 0x80000000u) : ~u;
  return __uint_as_float(b);
}

__global__ void init_attn(unsigned* __restrict__ amax, float* __restrict__ denom, int n4) {
  int i = blockIdx.x * blockDim.x + threadIdx.x;
  if (i < n4) { amax[i] = fenc(-3.0e38f); denom[i] = 0.f; }
}

__device__ __forceinline__ void edge_sd(const int* ei, int e, int E, int& s, int& d) {
  if (e < E) { s = ei[e]; d = ei[E + e]; }   // edge_index row0=src, row1=dst
  else       { s = d = e - E; }              // self-loops appended
}

__global__ void edge_max(const int* __restrict__ ei, const float* __restrict__ asr,
                         const float* __restrict__ adr, unsigned* __restrict__ amax,
                         int E, int Etot) {
  int e = blockIdx.x * blockDim.x + threadIdx.x;
  if (e >= Etot) return;
  int s, d; edge_sd(ei, e, E, s, d);
#pragma unroll
  for (int hh = 0; hh < HEADS; ++hh) {
    float v = asr[s * HEADS + hh] + adr[d * HEADS + hh];
    v = v > 0.f ? v : 0.2f * v;              // leaky_relu(0.2)
    atomicMax(&amax[d * HEADS + hh], fenc(v));
  }
}

__global__ void edge_exp(const int* __restrict__ ei, const float* __restrict__ asr,
                         const float* __restrict__ adr, const unsigned* __restrict__ amax,
                         float* __restrict__ ex, float* __restrict__ denom, int E, int Etot) {
  int e = blockIdx.x * blockDim.x + threadIdx.x;
  if (e >= Etot) return;
  int s, d; edge_sd(ei, e, E, s, d);
#pragma unroll
  for (int hh = 0; hh < HEADS; ++hh) {
    float v = asr[s * HEADS + hh] + adr[d * HEADS + hh];
    v = v > 0.f ? v : 0.2f * v;
    float m  = fdec(amax[d * HEADS + hh]);
    float ev = expf(v - m);
    ex[e * HEADS + hh] = ev;
    atomicAdd(&denom[d * HEADS + hh], ev);
  }
}

__global__ void init_bias(float* __restrict__ out, const float* __restrict__ b, int total) {
  int i = blockIdx.x * blockDim.x + threadIdx.x;
  if (i < total) out[i] = b[i & (DHID - 1)];
}

// One block (128 threads) per edge: out[d, :] += h[s, :] * coef(head)
__global__ void edge_scatter(const int* __restrict__ ei, const float* __restrict__ h,
                             const float* __restrict__ ex, const float* __restrict__ denom,
                             float* __restrict__ out, int E, int Etot) {
  int e = blockIdx.x;
  if (e >= Etot) return;
  int s, d; edge_sd(ei, e, E, s, d);
  int c  = threadIdx.x * 4;                  // 128 threads * 4 ch = 512
  int hh = c >> 7;                           // head = c / HID
  float coef = ex[e * HEADS + hh] / (denom[d * HEADS + hh] + 1e-16f);
  float4 hv = *(const float4*)(h + (size_t)s * DHID + c);
  float* o = out + (size_t)d * DHID + c;
  atomicAdd(o + 0, hv.x * coef);
  atomicAdd(o + 1, hv.y * coef);
  atomicAdd(o + 2, hv.z * coef);
  atomicAdd(o + 3, hv.w * coef);
}

// y = relu(y); if res: y = relu(y + res)   (matches nested relu in reference)
__global__ void relu_res(float* __restrict__ y, const float* __restrict__ res, int n) {
  int i = blockIdx.x * blockDim.x + threadIdx.x;
  if (i >= n) return;
  float t = y[i];
  t = t > 0.f ? t : 0.f;
  if (res) { t += res[i]; t = t > 0.f ? t : 0.f; }
  y[i] = t;
}

__global__ void zero_f32(float* __restrict__ p, int n) {
  int i = blockIdx.x * blockDim.x + threadIdx.x;
  if (i < n) p[i] = 0.f;
}

__global__ void copy_f32(float* __restrict__ dst, const float* __restrict__ src, int n) {
  int i = blockIdx.x * blockDim.x + threadIdx.x;
  if (i < n) dst[i] = src[i];
}

// BN stats: per-channel sum & sumsq with thread-local accumulation + one atomic
__global__ void bn_stats(const float* __restrict__ y, float* __restrict__ sums,
                         float* __restrict__ sq, int N) {
  int c0 = threadIdx.x * 2;                  // 256 threads * 2 ch = 512
  float s0 = 0, s1 = 0, q0 = 0, q1 = 0;
  for (int r = blockIdx.x; r < N; r += gridDim.x) {
    float2 v = *(const float2*)(y + (size_t)r * DHID + c0);
    s0 += v.x; q0 += v.x * v.x;
    s1 += v.y; q1 += v.y * v.y;
  }
  atomicAdd(&sums[c0], s0);     atomicAdd(&sums[c0 + 1], s1);
  atomicAdd(&sq[c0], q0);       atomicAdd(&sq[c0 + 1], q1);
}

__global__ void bn_apply(const float* __restrict__ y, const float* __restrict__ sums,
                         const float* __restrict__ sq, const float* __restrict__ gamma,
                         const float* __restrict__ beta, float* __restrict__ xo, int N) {
  int i = blockIdx.x * blockDim.x + threadIdx.x;
  if (i >= N * DHID) return;
  int c = i & (DHID - 1);
  float inv = 1.0f / (float)N;
  float m   = sums[c] * inv;
  float var = sq[c] * inv - m * m;
  xo[i] = gamma[c] * (y[i] - m) * rsqrtf(var + 1e-5f) + beta[c];
}

__global__ void zero_pool(float* __restrict__ pooled, int* __restrict__ cnt) {
  int i = blockIdx.x * blockDim.x + threadIdx.x;
  if (i < NGRAPH * DHID) pooled[i] = 0.f;
  if (i < NGRAPH) cnt[i] = 0;
}

__global__ void pool_sum(const float* __restrict__ x, const int* __restrict__ batch,
                         float* __restrict__ pooled, int* __restrict__ cnt, int N) {
  int n = blockIdx.x;
  if (n >= N) return;
  int g = batch[n];
  int c = threadIdx.x * 4;
  float4 v = *(const float4*)(x + (size_t)n * DHID + c);
  float* p = pooled + (size_t)g * DHID + c;
  atomicAdd(p + 0, v.x); atomicAdd(p + 1, v.y);
  atomicAdd(p + 2, v.z); atomicAdd(p + 3, v.w);
  if (threadIdx.x == 0) atomicAdd(&cnt[g], 1);
}

__global__ void pool_div(float* __restrict__ pooled, const int* __restrict__ cnt) {
  int i = blockIdx.x * blockDim.x + threadIdx.x;
  if (i >= NGRAPH * DHID) return;
  float c = (float)cnt[i >> 9];              // DHID = 512 = 1<<9
  pooled[i] /= (c < 1.f ? 1.f : c);
}

// ---------------------------------------------------------------------------
// Host side
// ---------------------------------------------------------------------------
static void gat_conv(const float* x, int in_dim, const float* W, const float* att_s,
                     const float* att_d, const float* bvec, float* hbuf, float* agg,
                     float* a_src, float* a_dst, unsigned* amax, float* denom, float* ex,
                     const int* ei, int N, int E, hipStream_t stream) {
  int Etot = E + N;
  dim3 gg((N + MBLK - 1) / MBLK, DHID / NBLK);
  gemm_wmma_tdm<<<gg, 256, 0, stream>>>(x, W, hbuf, N, in_dim, DHID);
  attn_scores<<<N, 128, 0, stream>>>(hbuf, att_s, att_d, a_src, a_dst);
  init_attn<<<(N * HEADS + 255) / 256, 256, 0, stream>>>(amax, denom, N * HEADS);
  edge_max<<<(Etot + 255) / 256, 256, 0, stream>>>(ei, a_src, a_dst, amax, E, Etot);
  edge_exp<<<(Etot + 255) / 256, 256, 0, stream>>>(ei, a_src, a_dst, amax, ex, denom, E, Etot);
  init_bias<<<((size_t)N * DHID + 255) / 256, 256, 0, stream>>>(agg, bvec, N * DHID);
  edge_scatter<<<Etot, 128, 0, stream>>>(ei, hbuf, ex, denom, agg, E, Etot);
}

extern "C" void kernel_launch(void* const* d_in, const int* in_sizes, int n_in,
                              void* d_out, int out_size, void* d_ws, size_t ws_size,
                              hipStream_t stream) {
  // Inputs in setup_inputs() insertion order, params flattened recursively:
  // 0:x 1:edge_index 2:batch 3:head.W 4:head.att_src 5:head.att_dst 6:head.b
  // 7+6i: layers[i].{gat.W, gat.att_src, gat.att_dst, gat.b, gamma, beta}
  // 31:out_w 32:out_b
  const float* x_in  = (const float*)d_in[0];
  const int*   ei    = (const int*)d_in[1];
  const int*   batch = (const int*)d_in[2];
  const int N = in_sizes[0] / IN_DIM;
  const int E = in_sizes[1] / 2;
  const int Etot = E + N;

  const float* headW  = (const float*)d_in[3];
  const float* headAs = (const float*)d_in[4];
  const float* headAd = (const float*)d_in[5];
  const float* headB  = (const float*)d_in[6];
  const float *layW[NLAYERS], *layAs[NLAYERS], *layAd[NLAYERS],
              *layB[NLAYERS], *layG[NLAYERS], *layBeta[NLAYERS];
  for (int i = 0; i < NLAYERS; ++i) {
    int b = 7 + 6 * i;
    layW[i]    = (const float*)d_in[b + 0];
    layAs[i]   = (const float*)d_in[b + 1];
    layAd[i]   = (const float*)d_in[b + 2];
    layB[i]    = (const float*)d_in[b + 3];
    layG[i]    = (const float*)d_in[b + 4];
    layBeta[i] = (const float*)d_in[b + 5];
  }
  const float* outW = (const float*)d_in[7 + 6 * NLAYERS];
  const float* outB = (const float*)d_in[8 + 6 * NLAYERS];

  // Carve workspace (256B aligned chunks)
  char* w = (char*)d_ws;
  auto take = [&](size_t bytes) -> void* {
    void* p = (void*)w;
    w += (bytes + 255) & ~(size_t)255;
    return p;
  };
  float*    xcur  = (float*)take((size_t)N * DHID * 4);
  float*    hbuf  = (float*)take((size_t)N * DHID * 4);
  float*    agg   = (float*)take((size_t)N * DHID * 4);
  float*    x0    = (float*)take((size_t)N * DHID * 4);
  float*    a_src = (float*)take((size_t)N * HEADS * 4);
  float*    a_dst = (float*)take((size_t)N * HEADS * 4);
  unsigned* amax  = (unsigned*)take((size_t)N * HEADS * 4);
  float*    denom = (float*)take((size_t)N * HEADS * 4);
  float*    ex    = (float*)take((size_t)Etot * HEADS * 4);
  float*    sums  = (float*)take((size_t)2 * DHID * 4);   // [sums | sumsq]
  float*    sq    = sums + DHID;
  float*    pooled= (float*)take((size_t)NGRAPH * DHID * 4);
  int*      cnt   = (int*)take((size_t)NGRAPH * 4);

  const int nfeat = N * DHID;

  // Head conv: xcur = gat(x_in)   (no relu/BN)
  gat_conv(x_in, IN_DIM, headW, headAs, headAd, headB, hbuf, xcur,
           a_src, a_dst, amax, denom, ex, ei, N, E, stream);

  for (int i = 0; i < NLAYERS; ++i) {
    if (i % 3 == 0)  // STEP = 3
      copy_f32<<<(nfeat + 255) / 256, 256, 0, stream>>>(x0, xcur, nfeat);

    gat_conv(xcur, DHID, layW[i], layAs[i], layAd[i], layB[i], hbuf, agg,
             a_src, a_dst, amax, denom, ex, ei, N, E, stream);

    // Residual schedule for NUM_LAYERS=4, STEP=3:
    // i=0: none; i=1: +x_in; i=2: +x_0; i=3 (last): +x_0(==x_in)
    const float* res = nullptr;
    if (i == 1 || i == 3) res = xcur;   // x_in (pre-layer activations, still intact)
    else if (i == 2)      res = x0;
    relu_res<<<(nfeat + 255) / 256, 256, 0, stream>>>(agg, res, nfeat);

    zero_f32<<<(2 * DHID + 255) / 256, 256, 0, stream>>>(sums, 2 * DHID);
    bn_stats<<<128, 256, 0, stream>>>(agg, sums, sq, N);
    bn_apply<<<(nfeat + 255) / 256, 256, 0, stream>>>(agg, sums, sq, layG[i], layBeta[i], xcur, N);
  }

  // Global mean pool + output projection
  zero_pool<<<(NGRAPH * DHID + 255) / 256, 256, 0, stream>>>(pooled, cnt);
  pool_sum<<<N, 128, 0, stream>>>(xcur, batch, pooled, cnt, N);
  pool_div<<<(NGRAPH * DHID + 255) / 256, 256, 0, stream>>>(pooled, cnt);

  dim3 gf((NGRAPH + 15) / 16, OUTD / 64);
  gemm_wmma_f32<<<gf, 32, 0, stream>>>(pooled, outW, outB, (float*)d_out, NGRAPH, DHID, OUTD);
}